// MultiBoxLoss_49787260895395
// MI455X (gfx1250) — compile-verified
//
#include <hip/hip_runtime.h>
#include <math.h>

#ifndef __has_builtin
#define __has_builtin(x) 0
#endif
#if __has_builtin(__builtin_amdgcn_global_load_async_to_lds_b32) && \
    __has_builtin(__builtin_amdgcn_s_wait_asynccnt)
#define USE_ASYNC_LDS 1
#else
#define USE_ASYNC_LDS 0
#endif

#define NB 32
#define NP 45180
#define NC 21
#define BLK1 256
#define NBX ((NP + BLK1 - 1) / BLK1) /* 177 */
#define TIE_CAP 4096

static __device__ __forceinline__ float clamp01(float v) {
  return fminf(fmaxf(v, 0.f), 1.f);
}

// SSD prior for flat index p (fidelity to reference make_priors()).
static __device__ void make_prior(int p, float& cx, float& cy, float& w, float& h) {
  int k, lp;
  if (p < 33750)      { k = 0; lp = p; }
  else if (p < 42414) { k = 1; lp = p - 33750; }
  else if (p < 44580) { k = 2; lp = p - 42414; }
  else                { k = 3; lp = p - 44580; }
  int cell = lp / 6, slot = lp - cell * 6;
  int f; float scale, s, s2, r0, r1;
  if (k == 0)      { f = 75; scale = 75.f;   s = 0.12f;        s2 = 0.2f;         r0 = 1.41421356f; r1 = 1.73205081f; }
  else if (k == 1) { f = 38; scale = 37.5f;  s = 0.33333334f;  s2 = 0.42031734f;  r0 = 2.0f;        r1 = 1.73205081f; }
  else if (k == 2) { f = 19; scale = 18.75f; s = 0.53f;        s2 = 0.66855568f;  r0 = 1.73205081f; r1 = 1.41421356f; }
  else             { f = 10; scale = 10.f;   s = 0.84333336f;  s2 = 0.91833186f;  r0 = 1.0f;        r1 = 1.0f; }
  int i = cell / f, j = cell - i * f;
  cx = ((float)j + 0.5f) / scale;
  cy = ((float)i + 0.5f) / scale;
  switch (slot) {
    case 0:  w = s;      h = s;      break;
    case 1:  w = s2;     h = s2;     break;
    case 2:  w = s * r0; h = s / r0; break;
    case 3:  w = s / r0; h = s * r0; break;
    case 4:  w = s * r1; h = s / r1; break;
    default: w = s / r1; h = s * r1; break;
  }
  cx = clamp01(cx); cy = clamp01(cy); w = clamp01(w); h = clamp01(h);
}

// ---------------- kernel 0: zero the integer accumulators -------------------
__global__ void mbl_zero(unsigned* __restrict__ numpos) {
  if (threadIdx.x < NB) numpos[threadIdx.x] = 0u;
}

// ---- kernel 1: log-softmax stats, keys/ce, positive count, CIoU partials ---
__global__ void __launch_bounds__(BLK1) mbl_main(
    const float*  __restrict__ conf,
    const float4* __restrict__ ploc,
    const int*    __restrict__ labels,
    const float4* __restrict__ gt,
    unsigned*     __restrict__ keys,
    float*        __restrict__ ce,
    unsigned*     __restrict__ numpos,
    float*        __restrict__ iou_part) {
  __shared__ float    redf[BLK1];
  __shared__ unsigned redu[BLK1];
#if USE_ASYNC_LDS
  __shared__ float stage[(BLK1 / 32) * NC * 32];
#endif
  const int tid = threadIdx.x;
  const int b   = blockIdx.y;
  const int p   = blockIdx.x * BLK1 + tid;

  float x[NC];
#if USE_ASYNC_LDS
  {
    // CDNA5 async global->LDS staging: each wave copies its 32 contiguous
    // rows (32*21 floats) flat into LDS, then gathers per-lane (bank-conflict
    // free since gcd(21,64)==1). Tracked by ASYNCcnt, wave-private.
    const int lane = tid & 31, wv = tid >> 5;
    const int pw0  = blockIdx.x * BLK1 + wv * 32;
    int rem    = NP - pw0;
    int vcount = rem > 0 ? (rem * NC < NC * 32 ? rem * NC : NC * 32) : 0;
    const float* gsrc  = conf + ((long long)b * NP + (long long)(pw0 > 0 ? pw0 : 0)) * NC;
    float*       lbase = &stage[wv * NC * 32];
#pragma unroll
    for (int c0 = 0; c0 < NC; ++c0) {
      int e = c0 * 32 + lane;
      if (e < vcount) {
        __builtin_amdgcn_global_load_async_to_lds_b32(
            (__attribute__((address_space(1))) int*)(gsrc + e),
            (__attribute__((address_space(3))) int*)(lbase + e), 0, 0);
      }
    }
    __builtin_amdgcn_s_wait_asynccnt(0);
    if (p < NP) {
#pragma unroll
      for (int c = 0; c < NC; ++c) x[c] = lbase[lane * NC + c];
    }
  }
#else
  if (p < NP) {
    const float* src = conf + ((long long)b * NP + p) * NC;
    __builtin_prefetch(src + (long long)BLK1 * NC, 0, 1);  // global_prefetch_b8
#pragma unroll
    for (int c = 0; c < NC; ++c) x[c] = src[c];
  }
#endif

  float    iouc = 0.f;
  unsigned cnt  = 0u;
  if (p < NP) {
    float m = x[0];
#pragma unroll
    for (int c = 1; c < NC; ++c) m = fmaxf(m, x[c]);
    float ssum = 0.f;
#pragma unroll
    for (int c = 0; c < NC; ++c) ssum += expf(x[c] - m);
    float     mlse = m + logf(ssum);
    long long idx  = (long long)b * NP + p;
    int  lab = labels[idx];
    bool pos = lab > 0;
    // monotonic key: pos -> 0 (i.e. -inf), else float bits of bg_loss (>=0) + 1
    float bg  = mlse - x[0];
    keys[idx] = pos ? 0u : (__float_as_uint(bg) + 1u);
    float xl = x[0];
#pragma unroll
    for (int c = 1; c < NC; ++c) xl = (c == lab) ? x[c] : xl;  // avoid scratch
    ce[idx] = mlse - xl;

    if (pos) {
      cnt = 1u;
      float cx, cy, pwid, phei;
      make_prior(p, cx, cy, pwid, phei);
      float4 pl  = ploc[idx];
      float  pcx = cx + pl.x * 0.1f * pwid;
      float  pcy = cy + pl.y * 0.1f * phei;
      float  pw  = pwid * expf(pl.z * 0.2f);
      float  ph  = phei * expf(pl.w * 0.2f);
      float x11 = pcx - 0.5f * pw, y11 = pcy - 0.5f * ph;
      float x12 = pcx + 0.5f * pw, y12 = pcy + 0.5f * ph;
      float4 g = gt[idx];
      float x21 = g.x, y21 = g.y, x22 = g.z, y22 = g.w;
      float w1 = x12 - x11, h1 = y12 - y11, w2 = x22 - x21, h2 = y22 - y21;
      float area1 = w1 * h1, area2 = w2 * h2;
      float cx1 = (x11 + x12) * 0.5f, cy1 = (y11 + y12) * 0.5f;
      float cx2 = (x21 + x22) * 0.5f, cy2 = (y21 + y22) * 0.5f;
      float iw = fmaxf(fminf(x12, x22) - fmaxf(x11, x21), 0.f);
      float ih = fmaxf(fminf(y12, y22) - fmaxf(y11, y21), 0.f);
      float inter = iw * ih;
      float uni   = area1 + area2 - inter;
      float iou   = inter / uni;
      float dx = cx2 - cx1, dy = cy2 - cy1;
      float inter_diag = dx * dx + dy * dy;
      float ow = fmaxf(fmaxf(x12, x22) - fminf(x11, x21), 0.f);
      float oh = fmaxf(fmaxf(y12, y22) - fminf(y11, y21), 0.f);
      float outer_diag = ow * ow + oh * oh;
      float u  = inter_diag / outer_diag;
      float dv = atanf(w2 / h2) - atanf(w1 / h1);
      float v  = 0.40528473f * dv * dv;  // 4/pi^2
      float alpha = v / ((1.f - iou) + v);
      float ciou  = fminf(fmaxf(iou - (u + alpha * v), -1.f), 1.f);
      iouc = 1.f - ciou;
    }
  }

  // deterministic per-block tree reduction -> one partial slot per block
  redf[tid] = iouc;
  redu[tid] = cnt;
  __syncthreads();
  for (int off = BLK1 / 2; off > 0; off >>= 1) {
    if (tid < off) { redf[tid] += redf[tid + off]; redu[tid] += redu[tid + off]; }
    __syncthreads();
  }
  if (tid == 0) {
    iou_part[blockIdx.y * gridDim.x + blockIdx.x] = redf[0];
    if (redu[0]) atomicAdd(&numpos[b], redu[0]);
  }
}

static __device__ float block_reduce_1024(float v, float* redf, int tid) {
  __syncthreads();
  redf[tid] = v;
  __syncthreads();
  for (int off = 512; off > 0; off >>= 1) {
    if (tid < off) redf[tid] += redf[tid + off];
    __syncthreads();
  }
  float r = redf[0];
  __syncthreads();
  return r;
}

// --- kernel 2: per-row exact top-K (radix select, stable ties) + CE sum -----
__global__ void __launch_bounds__(1024) mbl_select(
    const unsigned* __restrict__ keys,
    const float*    __restrict__ ce,
    const unsigned* __restrict__ numpos,
    float*          __restrict__ cls_part) {
  __shared__ unsigned hist[2048];
  __shared__ float    redf[1024];
  __shared__ unsigned tieIdx[TIE_CAP];
  __shared__ unsigned bc[3];
  const int       tid = threadIdx.x;
  const int       b   = blockIdx.x;
  const unsigned  K   = numpos[b] * 3u;
  const unsigned* key = keys + (long long)b * NP;
  const float*    cer = ce + (long long)b * NP;
  float total;

  if (K >= (unsigned)NP) {  // every rank < K -> mask is all-true
    float a = 0.f;
    for (int p = tid; p < NP; p += 1024) a += cer[p];
    total = block_reduce_1024(a, redf, tid);
  } else if (K == 0u) {  // positives only
    float a = 0.f;
    for (int p = tid; p < NP; p += 1024)
      if (key[p] == 0u) a += cer[p];
    total = block_reduce_1024(a, redf, tid);
  } else {
    // 3-level radix select of the K-th largest 32-bit key (11+11+10 bits)
    unsigned prefix = 0u, remaining = K;
    for (int lvl = 0; lvl < 3; ++lvl) {
      const int shift = (lvl == 0) ? 21 : ((lvl == 1) ? 10 : 0);
      const int nbins = (lvl == 2) ? 1024 : 2048;
      for (int i = tid; i < 2048; i += 1024) hist[i] = 0u;
      __syncthreads();
      for (int p = tid; p < NP; p += 1024) {
        unsigned kk = key[p];
        bool match = (lvl == 0) ? true
                   : (lvl == 1) ? ((kk >> 21) == (prefix >> 21))
                                : ((kk >> 10) == (prefix >> 10));
        if (match) atomicAdd(&hist[(kk >> shift) & (unsigned)(nbins - 1)], 1u);
      }
      __syncthreads();
      if (tid == 0) {
        unsigned cum = 0u; int chosen = 0;
        for (int bin = nbins - 1; bin >= 0; --bin) {
          unsigned c = hist[bin];
          if (cum + c >= remaining) { chosen = bin; break; }
          cum += c;
        }
        bc[0] = prefix | ((unsigned)chosen << shift);
        bc[1] = remaining - cum;
      }
      __syncthreads();
      prefix = bc[0]; remaining = bc[1];
      __syncthreads();
    }
    const unsigned T = prefix;     // exact threshold key
    const unsigned r = remaining;  // #ties at T to take (in index order)
    if (tid == 0) bc[2] = 0u;
    __syncthreads();
    float a = 0.f, at = 0.f;
    for (int p = tid; p < NP; p += 1024) {
      unsigned kk = key[p];
      if (kk == 0u || kk > T) a += cer[p];  // positives OR strictly above
      else if (kk == T) {
        at += cer[p];
        unsigned slot = atomicAdd(&bc[2], 1u);
        if (slot < TIE_CAP) tieIdx[slot] = (unsigned)p;
      }
    }
    __syncthreads();
    const unsigned E  = bc[2];
    float          S1 = block_reduce_1024(a, redf, tid);
    float          S2 = block_reduce_1024(at, redf, tid);
    if (E <= r) {
      total = S1 + S2;  // common case: all ties taken
    } else if (E <= TIE_CAP) {
      float accT = 0.f;  // take the r smallest indices among E ties
      for (unsigned e = (unsigned)tid; e < E; e += 1024u) {
        unsigned my = tieIdx[e], rank = 0u;
        for (unsigned q = 0; q < E; ++q) rank += (tieIdx[q] < my) ? 1u : 0u;
        if (rank < r) accT += cer[my];
      }
      total = S1 + block_reduce_1024(accT, redf, tid);
    } else {  // pathological duplicate flood: sequential exact fallback
      if (tid == 0) {
        float accT = 0.f; unsigned cntT = 0u;
        for (int p = 0; p < NP && cntT < r; ++p)
          if (key[p] == T) { accT += cer[p]; ++cntT; }
        redf[0] = S1 + accT;
      }
      __syncthreads();
      total = redf[0];
    }
  }
  if (tid == 0) cls_part[b] = total;
}

// ---------------- kernel 3: deterministic finalize --------------------------
__global__ void __launch_bounds__(256) mbl_fin(
    const unsigned* __restrict__ numpos,
    const float*    __restrict__ iou_part, int niou,
    const float*    __restrict__ cls_part,
    float*          __restrict__ out) {
  __shared__ float red[256];
  const int tid = threadIdx.x;
  float a = 0.f;
  for (int i = tid; i < niou; i += 256) a += iou_part[i];
  red[tid] = a;
  __syncthreads();
  for (int off = 128; off > 0; off >>= 1) {
    if (tid < off) red[tid] += red[tid + off];
    __syncthreads();
  }
  float iou_sum = red[0];
  __syncthreads();
  float c = (tid < NB) ? cls_part[tid] : 0.f;
  red[tid] = c;
  __syncthreads();
  for (int off = 128; off > 0; off >>= 1) {
    if (tid < off) red[tid] += red[tid + off];
    __syncthreads();
  }
  float cls_sum = red[0];
  if (tid == 0) {
    unsigned np = 0u;
    for (int bb = 0; bb < NB; ++bb) np += numpos[bb];
    float fnp = (float)np;
    out[0] = iou_sum / fnp;   // iou_loss / num_pos
    out[1] = cls_sum / fnp;   // classification_loss / num_pos
  }
}

extern "C" void kernel_launch(void* const* d_in, const int* in_sizes, int n_in,
                              void* d_out, int out_size, void* d_ws, size_t ws_size,
                              hipStream_t stream) {
  (void)in_sizes; (void)n_in; (void)out_size; (void)ws_size;
  const float*  conf   = (const float*)d_in[0];
  const float4* ploc   = (const float4*)d_in[1];
  const int*    labels = (const int*)d_in[2];
  const float4* gt     = (const float4*)d_in[3];
  float*        out    = (float*)d_out;

  char* ws = (char*)d_ws;
  const size_t BP = (size_t)NB * NP;
  unsigned* keys     = (unsigned*)ws;                       // BP * 4 bytes
  float*    ce       = (float*)(ws + BP * 4);               // BP * 4 bytes
  unsigned* numpos   = (unsigned*)(ws + BP * 8);            // 128 bytes
  float*    iou_part = (float*)(ws + BP * 8 + 256);         // NBX*NB * 4
  float*    cls_part = (float*)(ws + BP * 8 + 256 + (size_t)NBX * NB * 4);

  mbl_zero<<<1, 64, 0, stream>>>(numpos);
  mbl_main<<<dim3(NBX, NB), BLK1, 0, stream>>>(conf, ploc, labels, gt,
                                               keys, ce, numpos, iou_part);
  mbl_select<<<NB, 1024, 0, stream>>>(keys, ce, numpos, cls_part);
  mbl_fin<<<1, 256, 0, stream>>>(numpos, iou_part, NBX * NB, cls_part, out);
}